// Attention_63410897158465
// MI455X (gfx1250) — compile-verified
//
#include <hip/hip_runtime.h>

typedef float v2f __attribute__((ext_vector_type(2)));
typedef float v4f __attribute__((ext_vector_type(4)));
typedef float v8f __attribute__((ext_vector_type(8)));

#define NROW 512
#define DDIM 256
#define NPOS (NROW * NROW)

__device__ __forceinline__ float wave_sum(float x) {
#pragma unroll
    for (int off = 16; off > 0; off >>= 1)
        x += __shfl_xor(x, off, 32);
    return x;
}

// ---------------------------------------------------------------------------
// Phase 1: fused rank-1 projections. One wave32 per (i,j) position.
// Lane L handles elements [8L, 8L+8) of the D=256 dot product: two coalesced
// B128 loads per tensor (non-temporal: 768MB streamed exactly once), FMA with
// the weight vector (tiny, stays in cache), then a wave shuffle reduction.
// ---------------------------------------------------------------------------
__global__ __launch_bounds__(256) void proj_qkv_kernel(
    const float* __restrict__ xq, const float* __restrict__ xk,
    const float* __restrict__ xv,
    const float* __restrict__ wq, const float* __restrict__ wk,
    const float* __restrict__ wv,
    float* __restrict__ q, float* __restrict__ k, float* __restrict__ v)
{
    const int wave = (int)((blockIdx.x * blockDim.x + threadIdx.x) >> 5);
    const int lane = (int)(threadIdx.x & 31);
    const long base = (long)wave * DDIM + lane * 8;

    const v4f wq0 = *(const v4f*)(wq + lane * 8);
    const v4f wq1 = *(const v4f*)(wq + lane * 8 + 4);
    const v4f wk0 = *(const v4f*)(wk + lane * 8);
    const v4f wk1 = *(const v4f*)(wk + lane * 8 + 4);
    const v4f wv0 = *(const v4f*)(wv + lane * 8);
    const v4f wv1 = *(const v4f*)(wv + lane * 8 + 4);

    const v4f aq0 = __builtin_nontemporal_load((const v4f*)(xq + base));
    const v4f aq1 = __builtin_nontemporal_load((const v4f*)(xq + base) + 1);
    const v4f ak0 = __builtin_nontemporal_load((const v4f*)(xk + base));
    const v4f ak1 = __builtin_nontemporal_load((const v4f*)(xk + base) + 1);
    const v4f av0 = __builtin_nontemporal_load((const v4f*)(xv + base));
    const v4f av1 = __builtin_nontemporal_load((const v4f*)(xv + base) + 1);

    v4f pq = aq0 * wq0 + aq1 * wq1;
    v4f pk = ak0 * wk0 + ak1 * wk1;
    v4f pv = av0 * wv0 + av1 * wv1;

    float dq = wave_sum(pq.x + pq.y + pq.z + pq.w);
    float dk = wave_sum(pk.x + pk.y + pk.z + pk.w);
    float dv = wave_sum(pv.x + pv.y + pv.z + pv.w);

    if (lane == 0) {
        q[wave] = dq;
        k[wave] = dk;
        v[wave] = dv;
    }
}

// ---------------------------------------------------------------------------
// 512x512x512 fp32 matmul C = A*B (row-major, no transpose) using the native
// CDNA5 fp32 matrix op V_WMMA_F32_16X16X4_F32. One wave owns a 16x16 C tile,
// stepping K by 4. Operands are L2-resident (1MB each).
//
// Fragment layout per ISA 7.12.2 (wave32):
//   A 16x4 f32: lanes 0-15 M=lane K={0,1} in v0/v1; lanes 16-31 K={2,3}.
//   B 4x16 f32 mirrors with rows striped across lanes.
//   C/D: VGPR r, lanes 0-15 -> (M=r, N=lane); lanes 16-31 -> (M=r+8, N=lane-16).
// ---------------------------------------------------------------------------
__global__ __launch_bounds__(128) void mm512_wmma_f32(
    const float* __restrict__ A, const float* __restrict__ B,
    float* __restrict__ C)
{
    const int waveId = (int)((blockIdx.x * blockDim.x + threadIdx.x) >> 5);
    const int lane = (int)(threadIdx.x & 31);
    const int tm = waveId >> 5;   // 32x32 grid of 16x16 tiles
    const int tn = waveId & 31;
    const int l  = lane & 15;
    const int hi = lane >> 4;     // 0: K={0,1}/M=r ; 1: K={2,3}/M=r+8

    const float* arow = A + (long)(tm * 16 + l) * NROW + 2 * hi;
    const float* bcol = B + tn * 16 + l + (long)(2 * hi) * NROW;

    v8f c = {};
#pragma unroll 4
    for (int ks = 0; ks < NROW; ks += 4) {
        v2f a = *(const v2f*)(arow + ks);           // A[M, ks+2hi .. +1]
        v2f b;
        b.x = bcol[(long)ks * NROW];                // B[ks+2hi,   N]
        b.y = bcol[(long)ks * NROW + NROW];         // B[ks+2hi+1, N]
        c = __builtin_amdgcn_wmma_f32_16x16x4_f32(
            /*neg_a=*/false, a, /*neg_b=*/false, b,
            /*c_mod=*/(short)0, c, /*reuse_a=*/false, /*reuse_b=*/false);
    }

    float* crow = C + (long)(tm * 16 + hi * 8) * NROW + tn * 16 + l;
#pragma unroll
    for (int r = 0; r < 8; ++r)
        crow[(long)r * NROW] = c[r];
}

// ---------------------------------------------------------------------------
// Numerically-stable row softmax, in place. One 256-thread block per row,
// each thread owns 2 of the 512 elements.
// ---------------------------------------------------------------------------
__global__ __launch_bounds__(256) void softmax_rows(float* __restrict__ S)
{
    __shared__ float red[8];
    const int t = (int)threadIdx.x;
    float* srow = S + (long)blockIdx.x * NROW;

    float x0 = srow[t];
    float x1 = srow[t + 256];

    float m = fmaxf(x0, x1);
#pragma unroll
    for (int off = 16; off > 0; off >>= 1)
        m = fmaxf(m, __shfl_xor(m, off, 32));
    if ((t & 31) == 0) red[t >> 5] = m;
    __syncthreads();
    float bm = red[0];
#pragma unroll
    for (int i = 1; i < 8; ++i) bm = fmaxf(bm, red[i]);

    float e0 = expf(x0 - bm);
    float e1 = expf(x1 - bm);
    float s = wave_sum(e0 + e1);
    __syncthreads();
    if ((t & 31) == 0) red[t >> 5] = s;
    __syncthreads();
    float bs = 0.0f;
#pragma unroll
    for (int i = 0; i < 8; ++i) bs += red[i];

    const float inv = 1.0f / bs;
    srow[t]       = e0 * inv;
    srow[t + 256] = e1 * inv;
}

extern "C" void kernel_launch(void* const* d_in, const int* in_sizes, int n_in,
                              void* d_out, int out_size, void* d_ws, size_t ws_size,
                              hipStream_t stream) {
    const float* xq = (const float*)d_in[0];
    const float* xk = (const float*)d_in[1];
    const float* xv = (const float*)d_in[2];
    const float* wq = (const float*)d_in[3];
    const float* wk = (const float*)d_in[4];
    const float* wv = (const float*)d_in[5];

    float* ws = (float*)d_ws;
    float* q = ws;                 // 512*512 floats (1 MB)
    float* k = ws + NPOS;          // 1 MB
    float* v = ws + 2 * NPOS;      // 1 MB
    float* S = ws + 3 * NPOS;      // 1 MB scores / probabilities

    // Phase 1: one wave per (i,j) position; 262144 waves -> 32768 blocks x 256.
    proj_qkv_kernel<<<NPOS / 8, 256, 0, stream>>>(xq, xk, xv, wq, wk, wv, q, k, v);

    // Phase 2: S = q @ k  (1024 tiles, 4 waves/block -> 256 blocks x 128).
    mm512_wmma_f32<<<256, 128, 0, stream>>>(q, k, S);

    // Phase 3: softmax over last axis, in place.
    softmax_rows<<<NROW, 256, 0, stream>>>(S);

    // Phase 4: out = P @ v.
    mm512_wmma_f32<<<256, 128, 0, stream>>>(S, v, (float*)d_out);
}